// Model_2989297238407
// MI455X (gfx1250) — compile-verified
//
#include <hip/hip_runtime.h>
#include <cstdint>

#define N_USER   50000
#define N_ITEM   30000
#define N_ENTITY 100000
#define N_REL    20
#define DIM      128
#define N_EDGES  500000
#define ADJ_NNZ  2000000
#define N_UI     (N_USER + N_ITEM)
#define LEAK     0.2f
#define RES_LAMBDA 0.5f
#define EPSN     1e-12f
#define NEG_INF  (-3.402823466e38f)

typedef __attribute__((ext_vector_type(2))) float v2f;
typedef __attribute__((ext_vector_type(8))) float v8f;

// ---------------------------------------------------------------- utilities

__global__ void fill_f32(float* __restrict__ p, float v, long long n) {
  long long i = (long long)blockIdx.x * blockDim.x + threadIdx.x;
  if (i < n) p[i] = v;
}

__global__ void add_inplace(float* __restrict__ dst, const float* __restrict__ src, long long n) {
  long long i = (long long)blockIdx.x * blockDim.x + threadIdx.x;
  if (i < n) dst[i] += src[i];
}

__device__ inline float waveReduceSum(float s) {
#pragma unroll
  for (int off = 16; off > 0; off >>= 1) s += __shfl_xor(s, off, 32);
  return s;
}

__device__ inline void atomicMaxF(float* addr, float val) {
  int* ai = (int*)addr;
  int old = *ai;
  while (__int_as_float(old) < val) {
    int prev = atomicCAS(ai, old, __float_as_int(val));
    if (prev == old) break;
    old = prev;
  }
}

// ------------------------------------------------- entity projection (WMMA)
// Ph = ent @ W[0:128,:], Pt = ent @ W[128:256,:]   (ent: N_ENTITY x 128)
// Block = 256 threads = 8 waves; wave w owns output columns [16w, 16w+16).
// Each block covers 16 entity rows. Uses V_WMMA_F32_16X16X4_F32, K-loop of 32.
__global__ __launch_bounds__(256)
void kg_project_wmma(const float* __restrict__ ent, const float* __restrict__ W,
                     float* __restrict__ Ph, float* __restrict__ Pt) {
  const int wave  = threadIdx.x >> 5;       // 0..7 -> N tile
  const int lane  = threadIdx.x & 31;
  const int half  = lane >> 4;              // 0: lanes 0-15, 1: lanes 16-31
  const int m     = lane & 15;
  const int mbase = blockIdx.x * 16;
  const int ncol  = wave * 16 + m;

  v8f ch = {};
  v8f ct = {};
  const float* __restrict__ arow = ent + (size_t)(mbase + m) * DIM;

#pragma unroll 4
  for (int k = 0; k < DIM; k += 4) {
    const int ka = k + 2 * half;
    // A fragment: 16x4 f32 tile (ISA 7.12.2): lane half selects K pair
    v2f a;
    a.x = arow[ka];
    a.y = arow[ka + 1];
    // B fragments: 4x16 f32, row-major W; K rows striped like A across halves
    v2f bh, bt;
    bh.x = W[(size_t)ka * DIM + ncol];
    bh.y = W[(size_t)(ka + 1) * DIM + ncol];
    bt.x = W[(size_t)(ka + DIM) * DIM + ncol];
    bt.y = W[(size_t)(ka + 1 + DIM) * DIM + ncol];
    ch = __builtin_amdgcn_wmma_f32_16x16x4_f32(false, a, false, bh, (short)0, ch, false, false);
    ct = __builtin_amdgcn_wmma_f32_16x16x4_f32(false, a, false, bt, (short)0, ct, false, false);
  }

  // C/D layout: VGPR r -> (M = r + 8*half, N = lane&15)
#pragma unroll
  for (int r = 0; r < 8; ++r) {
    const int row = mbase + r + 8 * half;
    Ph[(size_t)row * DIM + ncol] = ch[r];
    Pt[(size_t)row * DIM + ncol] = ct[r];
  }
}

// ------------------------------------------------------- edge attention ops
// One wave32 per edge; lane l covers float4 slice [4l, 4l+4) of D=128.
__global__ __launch_bounds__(256)
void kg_edge_score(const float* __restrict__ Ph, const float* __restrict__ Pt,
                   const float* __restrict__ rEmb,
                   const int* __restrict__ head, const int* __restrict__ tail,
                   const int* __restrict__ etype,
                   float* __restrict__ escore, float* __restrict__ mmax) {
  const long long e = (long long)blockIdx.x * 8 + (threadIdx.x >> 5);
  if (e >= N_EDGES) return;
  const int lane = threadIdx.x & 31;
  const int h = head[e], t = tail[e], r = etype[e];
  const float4 a = ((const float4*)(Ph   + (size_t)h * DIM))[lane];
  const float4 b = ((const float4*)(Pt   + (size_t)t * DIM))[lane];
  const float4 c = ((const float4*)(rEmb + (size_t)r * DIM))[lane];
  float s = (a.x + b.x) * c.x + (a.y + b.y) * c.y +
            (a.z + b.z) * c.z + (a.w + b.w) * c.w;
  s = waveReduceSum(s);
  if (lane == 0) {
    s = s > 0.0f ? s : LEAK * s;            // leaky relu
    escore[e] = s;
    atomicMaxF(&mmax[h], s);                // segment_max
  }
}

__global__ void kg_edge_denom(const float* __restrict__ escore,
                              const int* __restrict__ head,
                              const float* __restrict__ mmax,
                              float* __restrict__ denom) {
  const long long e = (long long)blockIdx.x * blockDim.x + threadIdx.x;
  if (e >= N_EDGES) return;
  const int h = head[e];
  atomicAdd(&denom[h], __expf(escore[e] - mmax[h]));
}

__global__ __launch_bounds__(256)
void kg_edge_agg(const float* __restrict__ escore,
                 const int* __restrict__ head, const int* __restrict__ tail,
                 const float* __restrict__ mmax, const float* __restrict__ denom,
                 const float* __restrict__ ent, float* __restrict__ agg) {
  const long long e = (long long)blockIdx.x * 8 + (threadIdx.x >> 5);
  if (e >= N_EDGES) return;
  const int lane = threadIdx.x & 31;
  const int h = head[e], t = tail[e];
  const float alpha = __expf(escore[e] - mmax[h]) / denom[h];
  const float4 v = ((const float4*)(ent + (size_t)t * DIM))[lane];
  float* dst = agg + (size_t)h * DIM + lane * 4;
  atomicAdd(dst + 0, alpha * v.x);
  atomicAdd(dst + 1, alpha * v.y);
  atomicAdd(dst + 2, alpha * v.z);
  atomicAdd(dst + 3, alpha * v.w);
}

// new_ent = l2norm(agg + ent); res = 0.5*res + new_ent; ent = new_ent
__global__ __launch_bounds__(256)
void kg_norm_res(const float* __restrict__ agg, float* __restrict__ ent,
                 float* __restrict__ res) {
  const long long i = (long long)blockIdx.x * 8 + (threadIdx.x >> 5);
  if (i >= N_ENTITY) return;
  const int lane = threadIdx.x & 31;
  const float4 a = ((const float4*)(agg + (size_t)i * DIM))[lane];
  const float4 e = ((const float4*)(ent + (size_t)i * DIM))[lane];
  float vx = a.x + e.x, vy = a.y + e.y, vz = a.z + e.z, vw = a.w + e.w;
  const float ss = waveReduceSum(vx * vx + vy * vy + vz * vz + vw * vw);
  const float inv = 1.0f / fmaxf(sqrtf(ss), EPSN);
  vx *= inv; vy *= inv; vz *= inv; vw *= inv;
  float4* ep = (float4*)(ent + (size_t)i * DIM) + lane;
  float4* rp = (float4*)(res + (size_t)i * DIM) + lane;
  float4 ro = *rp;
  *ep = make_float4(vx, vy, vz, vw);
  *rp = make_float4(RES_LAMBDA * ro.x + vx, RES_LAMBDA * ro.y + vy,
                    RES_LAMBDA * ro.z + vz, RES_LAMBDA * ro.w + vw);
}

// ----------------------------------------------------------- GNN / SpMM part

__global__ void build_embeds(const float* __restrict__ u, const float* __restrict__ res,
                             float* __restrict__ total, float* __restrict__ cur) {
  const long long i = (long long)blockIdx.x * blockDim.x + threadIdx.x;
  if (i >= (long long)N_UI * DIM) return;
  const long long row = i / DIM;
  const float v = (row < N_USER) ? u[i] : res[i - (long long)N_USER * DIM];
  total[i] = v;
  cur[i] = v;
}

__global__ __launch_bounds__(256)
void spmm_atomic(const int* __restrict__ rows, const int* __restrict__ cols,
                 const float* __restrict__ vals, const float* __restrict__ x,
                 float* __restrict__ y) {
  const long long e = (long long)blockIdx.x * 8 + (threadIdx.x >> 5);
  if (e >= ADJ_NNZ) return;
  const int lane = threadIdx.x & 31;
  const int r = rows[e], c = cols[e];
  const float v = vals[e];
  const float4 xv = ((const float4*)(x + (size_t)c * DIM))[lane];
  float* dst = y + (size_t)r * DIM + lane * 4;
  atomicAdd(dst + 0, v * xv.x);
  atomicAdd(dst + 1, v * xv.y);
  atomicAdd(dst + 2, v * xv.z);
  atomicAdd(dst + 3, v * xv.w);
}

// ---------------------------------------------------------------- launcher

extern "C" void kernel_launch(void* const* d_in, const int* in_sizes, int n_in,
                              void* d_out, int out_size, void* d_ws, size_t ws_size,
                              hipStream_t stream) {
  const float* uEmb     = (const float*)d_in[0];
  const float* eEmb     = (const float*)d_in[1];
  const float* rEmb     = (const float*)d_in[2];
  const float* W        = (const float*)d_in[3];
  const float* adj_vals = (const float*)d_in[4];
  const int*   edge_idx = (const int*)d_in[5];
  const int*   etype    = (const int*)d_in[6];
  const int*   adj_rows = (const int*)d_in[7];
  const int*   adj_cols = (const int*)d_in[8];
  const int* head = edge_idx;
  const int* tail = edge_idx + N_EDGES;

  const size_t entB = (size_t)N_ENTITY * DIM * sizeof(float);
  char* p = (char*)d_ws;
  float* ent    = (float*)p; p += entB;
  float* res    = (float*)p; p += entB;
  float* Ph     = (float*)p; p += entB;   // later aliased: agg, then SpMM nxt
  float* Pt     = (float*)p; p += entB;   // later aliased: SpMM cur
  float* escore = (float*)p; p += (size_t)N_EDGES * sizeof(float);
  float* mmax   = (float*)p; p += (size_t)N_ENTITY * sizeof(float);
  float* denom  = (float*)p; p += (size_t)N_ENTITY * sizeof(float);

  float* total = (float*)d_out;           // concat(user rows, item rows) = whole total
  float* agg = Ph;

  hipMemcpyAsync(ent, eEmb, entB, hipMemcpyDeviceToDevice, stream);
  hipMemcpyAsync(res, eEmb, entB, hipMemcpyDeviceToDevice, stream);

  const long long nEnt = (long long)N_ENTITY * DIM;
  const long long nUI  = (long long)N_UI * DIM;

  for (int hop = 0; hop < 2; ++hop) {
    // 1) dense projection on tensor cores (f32 WMMA): Ph/Pt = ent @ Wh/Wt
    kg_project_wmma<<<N_ENTITY / 16, 256, 0, stream>>>(ent, W, Ph, Pt);
    // 2) segment-softmax scores
    fill_f32<<<(N_ENTITY + 255) / 256, 256, 0, stream>>>(mmax, NEG_INF, N_ENTITY);
    fill_f32<<<(N_ENTITY + 255) / 256, 256, 0, stream>>>(denom, 0.0f, N_ENTITY);
    kg_edge_score<<<(N_EDGES + 7) / 8, 256, 0, stream>>>(Ph, Pt, rEmb, head, tail,
                                                         etype, escore, mmax);
    kg_edge_denom<<<(N_EDGES + 255) / 256, 256, 0, stream>>>(escore, head, mmax, denom);
    // 3) weighted aggregation (agg aliases Ph — Ph dead after scoring)
    fill_f32<<<(int)((nEnt + 255) / 256), 256, 0, stream>>>(agg, 0.0f, nEnt);
    kg_edge_agg<<<(N_EDGES + 7) / 8, 256, 0, stream>>>(escore, head, tail, mmax,
                                                       denom, ent, agg);
    // 4) residual + L2 norm
    kg_norm_res<<<(N_ENTITY + 7) / 8, 256, 0, stream>>>(agg, ent, res);
  }

  // GNN phase: total lives in d_out; ping-pong buffers reuse Pt / Ph
  float* cur = Pt;
  float* nxt = Ph;
  build_embeds<<<(int)((nUI + 255) / 256), 256, 0, stream>>>(uEmb, res, total, cur);
  for (int layer = 0; layer < 2; ++layer) {
    fill_f32<<<(int)((nUI + 255) / 256), 256, 0, stream>>>(nxt, 0.0f, nUI);
    spmm_atomic<<<(ADJ_NNZ + 7) / 8, 256, 0, stream>>>(adj_rows, adj_cols, adj_vals,
                                                       cur, nxt);
    add_inplace<<<(int)((nUI + 255) / 256), 256, 0, stream>>>(total, nxt, nUI);
    float* t = cur; cur = nxt; nxt = t;
  }
}